// HalfKP_NNUE_37589553775220
// MI455X (gfx1250) — compile-verified
//
#include <hip/hip_runtime.h>
#include <hip/hip_bf16.h>

typedef __attribute__((ext_vector_type(2))) float v2f;
typedef __attribute__((ext_vector_type(4))) float v4f;
typedef __attribute__((ext_vector_type(8))) float v8f;

#define NUM_FEATURES 40960
#define KTILE        256
#define ROWS         16
#define LDS_STRIDE   260   // 256 + 4 pad -> stride % 64 == 4 -> conflict-free WMMA operand reads

__global__ __launch_bounds__(256) void nnue_fused_kernel(
    const float* __restrict__ wf,   // (B, 40960) white features
    const float* __restrict__ bf,   // (B, 40960) black features
    const float* __restrict__ stm,  // (B,)
    const float* __restrict__ ft_w, // (4, 40960)
    const float* __restrict__ ft_b, // (4,)
    const float* __restrict__ l1_w, // (8, 8)
    const float* __restrict__ l1_b, // (8,)
    const float* __restrict__ l2_w, // (1, 8)
    const float* __restrict__ l2_b, // (1,)
    float* __restrict__ out)        // (B, 1)
{
    __shared__ float sW[ROWS][LDS_STRIDE];   // white feature tile
    __shared__ float sB[ROWS][LDS_STRIDE];   // black feature tile
    __shared__ float sFt[16][LDS_STRIDE];    // ft_w tile; rows 4..15 stay zero (N padding)
    __shared__ float sRed[8][2][16][4];      // per-wave partial accumulators

    const int t    = threadIdx.x;
    const int wave = t >> 5;
    const int lane = t & 31;
    const int row_base = blockIdx.x * ROWS;

    // Zero the N-padding rows of the weight tile once (never rewritten).
    for (int i = t; i < 12 * LDS_STRIDE; i += 256)
        sFt[4 + i / LDS_STRIDE][i % LDS_STRIDE] = 0.0f;

    v8f cw = {};   // white accumulator, 16x16 f32 C tile (cols 0..3 useful)
    v8f cb = {};   // black accumulator

    // WMMA 32-bit A/B operand layout (wave32):
    // lanes 0..15  : row/col = lane,     VGPR0 = K+0, VGPR1 = K+1
    // lanes 16..31 : row/col = lane-16,  VGPR0 = K+2, VGPR1 = K+3
    const int lrow  = lane & 15;
    const int khalf = (lane >> 4) << 1;

    const int col = (t & 63) << 2;   // staging: 64 threads cover one 1KB row
    const int r0  = t >> 6;

    __syncthreads();

    for (int k0 = 0; k0 < NUM_FEATURES; k0 += KTILE) {
        // ---- stage tiles to LDS (coalesced b128 global loads, NT hint) ----
        #pragma unroll
        for (int r = 0; r < 4; ++r) {
            const int row = (r << 2) + r0;
            const size_t g = (size_t)(row_base + row) * NUM_FEATURES + k0 + col;
            v4f a = __builtin_nontemporal_load((const v4f*)(wf + g));
            v4f b = __builtin_nontemporal_load((const v4f*)(bf + g));
            *(v4f*)&sW[row][col] = a;
            *(v4f*)&sB[row][col] = b;
        }
        {
            const int wrow = r0;  // 4 weight rows x 64 chunks
            const size_t g = (size_t)wrow * NUM_FEATURES + k0 + col;
            *(v4f*)&sFt[wrow][col] = *(const v4f*)(ft_w + g);
        }
        __syncthreads();

        // ---- each wave: 32-wide K slice -> 8 WMMA steps per side ----
        #pragma unroll
        for (int s = 0; s < 8; ++s) {
            const int kk = (wave << 5) + (s << 2) + khalf;
            v2f aw = *(const v2f*)&sW[lrow][kk];
            v2f ab = *(const v2f*)&sB[lrow][kk];
            v2f wv = *(const v2f*)&sFt[lrow][kk];
            cw = __builtin_amdgcn_wmma_f32_16x16x4_f32(false, aw, false, wv,
                                                       (short)0, cw, false, false);
            cb = __builtin_amdgcn_wmma_f32_16x16x4_f32(false, ab, false, wv,
                                                       (short)0, cb, false, false);
        }
        __syncthreads();
    }

    // ---- cross-wave reduction of partial C tiles (only N<4 columns matter) ----
    {
        const int n = lane & 15;
        if (n < 4) {
            const int mbase = (lane >> 4) << 3;   // lanes>=16 hold M=8+v
            #pragma unroll
            for (int v = 0; v < 8; ++v) {
                sRed[wave][0][mbase + v][n] = cw[v];
                sRed[wave][1][mbase + v][n] = cb[v];
            }
        }
    }
    __syncthreads();

    // ---- fused tail: bias + stm blend + clip + L1 + clip + L2 ----
    if (t < 16) {
        const int m = t;
        float w[4] = {0.f, 0.f, 0.f, 0.f};
        float b[4] = {0.f, 0.f, 0.f, 0.f};
        #pragma unroll
        for (int wv = 0; wv < 8; ++wv)
            #pragma unroll
            for (int j = 0; j < 4; ++j) {
                w[j] += sRed[wv][0][m][j];
                b[j] += sRed[wv][1][m][j];
            }
        const float s = stm[row_base + m];
        float acc[8];
        #pragma unroll
        for (int j = 0; j < 4; ++j) {
            const float wj = w[j] + ft_b[j];
            const float bj = b[j] + ft_b[j];
            acc[j]     = s * wj + (1.0f - s) * bj;
            acc[4 + j] = s * bj + (1.0f - s) * wj;
        }
        #pragma unroll
        for (int j = 0; j < 8; ++j)
            acc[j] = fminf(fmaxf(acc[j], 0.0f), 1.0f);
        float o = l2_b[0];
        #pragma unroll
        for (int i = 0; i < 8; ++i) {
            float h = l1_b[i];
            #pragma unroll
            for (int j = 0; j < 8; ++j)
                h += l1_w[i * 8 + j] * acc[j];
            h = fminf(fmaxf(h, 0.0f), 1.0f);
            o += l2_w[i] * h;
        }
        out[row_base + m] = o;
    }
}

extern "C" void kernel_launch(void* const* d_in, const int* in_sizes, int n_in,
                              void* d_out, int out_size, void* d_ws, size_t ws_size,
                              hipStream_t stream) {
    const float* wf   = (const float*)d_in[0];
    const float* bf   = (const float*)d_in[1];
    const float* stm  = (const float*)d_in[2];
    const float* ft_w = (const float*)d_in[3];
    const float* ft_b = (const float*)d_in[4];
    const float* l1_w = (const float*)d_in[5];
    const float* l1_b = (const float*)d_in[6];
    const float* l2_w = (const float*)d_in[7];
    const float* l2_b = (const float*)d_in[8];
    float* out = (float*)d_out;

    const int B = in_sizes[2];          // batch = stm element count (8192)
    const int grid = B / ROWS;          // 16 rows per workgroup

    nnue_fused_kernel<<<grid, 256, 0, stream>>>(wf, bf, stm, ft_w, ft_b,
                                                l1_w, l1_b, l2_w, l2_b, out);
}